// TemporalGraphSAGE_21732534518206
// MI455X (gfx1250) — compile-verified
//
#include <hip/hip_runtime.h>
#include <hip/hip_bf16.h>
#include <math.h>

#define N_NODES 100000
#define N_EDGES 500000
#define FEAT    128
#define HID     256
#define N_LINKS 16384

typedef float v2f __attribute__((ext_vector_type(2)));
typedef float v8f __attribute__((ext_vector_type(8)));

// ---------------------------------------------------------------------------
// Utility kernels
// ---------------------------------------------------------------------------
__global__ __launch_bounds__(256) void zero_kernel(float* __restrict__ p, int n) {
    int i = blockIdx.x * blockDim.x + threadIdx.x;
    if (i < n) p[i] = 0.0f;
}

// in-degree: deg[dst[e]] += 1
__global__ __launch_bounds__(256) void degree_kernel(const int* __restrict__ ei,
                                                     float* __restrict__ deg) {
    int e = blockIdx.x * blockDim.x + threadIdx.x;
    if (e < N_EDGES) atomicAdd(&deg[ei[N_EDGES + e]], 1.0f);
}

// nb[dst[e]][:] += h[src[e]][:]   (one thread = one edge x 4 channels)
__global__ __launch_bounds__(256) void scatter_kernel(const int* __restrict__ ei,
                                                      const float* __restrict__ h,
                                                      float* __restrict__ nb) {
    int i = blockIdx.x * blockDim.x + threadIdx.x;
    if (i >= N_EDGES * (HID / 4)) return;
    int e = i >> 6;
    int c = (i & 63) * 4;
    int s = ei[e];
    int d = ei[N_EDGES + e];
    const float4 v = *reinterpret_cast<const float4*>(h + (size_t)s * HID + c);
    float* base = nb + (size_t)d * HID + c;
    atomicAdd(base + 0, v.x);
    atomicAdd(base + 1, v.y);
    atomicAdd(base + 2, v.z);
    atomicAdd(base + 3, v.w);
}

// nb = (nb + h) / (deg + 1)      (in place)
__global__ __launch_bounds__(256) void mean_kernel(float* __restrict__ nb,
                                                   const float* __restrict__ h,
                                                   const float* __restrict__ deg) {
    int i = blockIdx.x * blockDim.x + threadIdx.x;
    if (i >= N_NODES * (HID / 4)) return;
    int node = i >> 6;
    int c = (i & 63) * 4;
    float inv = 1.0f / (deg[node] + 1.0f);
    float4 a = *reinterpret_cast<const float4*>(nb + (size_t)node * HID + c);
    float4 b = *reinterpret_cast<const float4*>(h + (size_t)node * HID + c);
    float4 r;
    r.x = (a.x + b.x) * inv;
    r.y = (a.y + b.y) * inv;
    r.z = (a.z + b.z) * inv;
    r.w = (a.w + b.w) * inv;
    *reinterpret_cast<float4*>(nb + (size_t)node * HID + c) = r;
}

// combined[m] = [s, d, s*d, |s-d|]   (4*HID wide)
__global__ __launch_bounds__(256) void combined_kernel(const float* __restrict__ h,
                                                       const int* __restrict__ src_idx,
                                                       const int* __restrict__ dst_idx,
                                                       float* __restrict__ comb) {
    int i = blockIdx.x * blockDim.x + threadIdx.x;
    if (i >= N_LINKS * (HID / 4)) return;
    int m = i >> 6;
    int c = (i & 63) * 4;
    float4 s = *reinterpret_cast<const float4*>(h + (size_t)src_idx[m] * HID + c);
    float4 d = *reinterpret_cast<const float4*>(h + (size_t)dst_idx[m] * HID + c);
    float* row = comb + (size_t)m * (4 * HID);
    *reinterpret_cast<float4*>(row + c) = s;
    *reinterpret_cast<float4*>(row + HID + c) = d;
    float4 p;
    p.x = s.x * d.x; p.y = s.y * d.y; p.z = s.z * d.z; p.w = s.w * d.w;
    *reinterpret_cast<float4*>(row + 2 * HID + c) = p;
    float4 q;
    q.x = fabsf(s.x - d.x); q.y = fabsf(s.y - d.y);
    q.z = fabsf(s.z - d.z); q.w = fabsf(s.w - d.w);
    *reinterpret_cast<float4*>(row + 3 * HID + c) = q;
}

// ---------------------------------------------------------------------------
// FP32 WMMA GEMM:  D[M,N] = A[M,K] * W[N,K]^T + bias  (optional ReLU)
// 256 threads = 8 waves (2M x 4N of 16x16 tiles) -> 32x64 block tile.
// Register double-buffering: chunk k+1 is fetched into VGPRs while WMMA
// consumes chunk k from LDS, hiding global-load latency behind compute.
// Requires M%32==0, N%64==0, K%32==0 (true at all call sites).
// ---------------------------------------------------------------------------
template <int RELU>
__global__ __launch_bounds__(256)
void wmma_gemm_kernel(const float* __restrict__ A, const float* __restrict__ W,
                      const float* __restrict__ bias, float* __restrict__ D,
                      int M, int N, int K) {
    __shared__ float As[32][36];   // +4 pad: 16B-aligned rows, conflict-free frags
    __shared__ float Ws[64][36];

    const int t    = threadIdx.x;
    const int lane = t & 31;
    const int wave = t >> 5;
    const int mw   = wave & 1;
    const int nw   = wave >> 1;
    const int lm   = lane & 15;
    const int lkhi = (lane >> 4) << 1;  // K sub-offset: 0 or 2
    const int blockM = blockIdx.x * 32;
    const int blockN = blockIdx.y * 64;

    // per-thread staging slots (coalesced float4 = global_load_b128)
    const int r = t >> 3, c = (t & 7) * 4;
    const float* Arow  = A + (size_t)(blockM + r) * K + c;
    const float* Wrow0 = W + (size_t)(blockN + r) * K + c;
    const float* Wrow1 = W + (size_t)(blockN + 32 + r) * K + c;

    v8f acc = {};
    float4 qa  = *reinterpret_cast<const float4*>(Arow);
    float4 qw0 = *reinterpret_cast<const float4*>(Wrow0);
    float4 qw1 = *reinterpret_cast<const float4*>(Wrow1);

    for (int k0 = 0; k0 < K; k0 += 32) {
        *reinterpret_cast<float4*>(&As[r][c])      = qa;
        *reinterpret_cast<float4*>(&Ws[r][c])      = qw0;
        *reinterpret_cast<float4*>(&Ws[r + 32][c]) = qw1;
        __syncthreads();
        if (k0 + 32 < K) {   // prefetch next chunk; completes during WMMA phase
            qa  = *reinterpret_cast<const float4*>(Arow + k0 + 32);
            qw0 = *reinterpret_cast<const float4*>(Wrow0 + k0 + 32);
            qw1 = *reinterpret_cast<const float4*>(Wrow1 + k0 + 32);
        }
#pragma unroll
        for (int kk = 0; kk < 8; ++kk) {
            int kb = kk * 4 + lkhi;
            v2f a, b;
            a.x = As[mw * 16 + lm][kb];
            a.y = As[mw * 16 + lm][kb + 1];
            b.x = Ws[nw * 16 + lm][kb];
            b.y = Ws[nw * 16 + lm][kb + 1];
            acc = __builtin_amdgcn_wmma_f32_16x16x4_f32(
                false, a, false, b, (short)0, acc, false, false);
        }
        __syncthreads();
    }

    const int n  = blockN + nw * 16 + lm;
    const float bv = bias ? bias[n] : 0.0f;
    const int m0 = blockM + mw * 16 + ((lane >> 4) << 3);
#pragma unroll
    for (int v = 0; v < 8; ++v) {
        float val = acc[v] + bv;
        if (RELU) val = fmaxf(val, 0.0f);
        D[(size_t)(m0 + v) * N + n] = val;
    }
}

// ---------------------------------------------------------------------------
// Fused SAGE-layer GEMM:
//   D = ReLU( A0 @ W0^T + A1 @ W1^T + bias )
// Both products accumulate into the same WMMA accumulator inside the K-loop
// (16 wmma per chunk per wave) -- avoids a second pass that would re-read and
// re-write the 100 MB output. LDS: 4 staged tiles = 27 KB.
// ---------------------------------------------------------------------------
template <int RELU>
__global__ __launch_bounds__(256)
void wmma_dual_gemm_kernel(const float* __restrict__ A0, const float* __restrict__ A1,
                           const float* __restrict__ W0, const float* __restrict__ W1,
                           const float* __restrict__ bias, float* __restrict__ D,
                           int M, int N, int K) {
    __shared__ float As0[32][36];
    __shared__ float As1[32][36];
    __shared__ float Ws0[64][36];
    __shared__ float Ws1[64][36];

    const int t    = threadIdx.x;
    const int lane = t & 31;
    const int wave = t >> 5;
    const int mw   = wave & 1;
    const int nw   = wave >> 1;
    const int lm   = lane & 15;
    const int lkhi = (lane >> 4) << 1;
    const int blockM = blockIdx.x * 32;
    const int blockN = blockIdx.y * 64;

    const int r = t >> 3, c = (t & 7) * 4;
    const float* A0row  = A0 + (size_t)(blockM + r) * K + c;
    const float* A1row  = A1 + (size_t)(blockM + r) * K + c;
    const float* W0row0 = W0 + (size_t)(blockN + r) * K + c;
    const float* W0row1 = W0 + (size_t)(blockN + 32 + r) * K + c;
    const float* W1row0 = W1 + (size_t)(blockN + r) * K + c;
    const float* W1row1 = W1 + (size_t)(blockN + 32 + r) * K + c;

    v8f acc = {};
    float4 qa0  = *reinterpret_cast<const float4*>(A0row);
    float4 qa1  = *reinterpret_cast<const float4*>(A1row);
    float4 qw00 = *reinterpret_cast<const float4*>(W0row0);
    float4 qw01 = *reinterpret_cast<const float4*>(W0row1);
    float4 qw10 = *reinterpret_cast<const float4*>(W1row0);
    float4 qw11 = *reinterpret_cast<const float4*>(W1row1);

    for (int k0 = 0; k0 < K; k0 += 32) {
        *reinterpret_cast<float4*>(&As0[r][c])      = qa0;
        *reinterpret_cast<float4*>(&As1[r][c])      = qa1;
        *reinterpret_cast<float4*>(&Ws0[r][c])      = qw00;
        *reinterpret_cast<float4*>(&Ws0[r + 32][c]) = qw01;
        *reinterpret_cast<float4*>(&Ws1[r][c])      = qw10;
        *reinterpret_cast<float4*>(&Ws1[r + 32][c]) = qw11;
        __syncthreads();
        if (k0 + 32 < K) {   // prefetch next chunk behind the WMMA phase
            qa0  = *reinterpret_cast<const float4*>(A0row + k0 + 32);
            qa1  = *reinterpret_cast<const float4*>(A1row + k0 + 32);
            qw00 = *reinterpret_cast<const float4*>(W0row0 + k0 + 32);
            qw01 = *reinterpret_cast<const float4*>(W0row1 + k0 + 32);
            qw10 = *reinterpret_cast<const float4*>(W1row0 + k0 + 32);
            qw11 = *reinterpret_cast<const float4*>(W1row1 + k0 + 32);
        }
#pragma unroll
        for (int kk = 0; kk < 8; ++kk) {
            int kb = kk * 4 + lkhi;
            v2f a, b;
            a.x = As0[mw * 16 + lm][kb];
            a.y = As0[mw * 16 + lm][kb + 1];
            b.x = Ws0[nw * 16 + lm][kb];
            b.y = Ws0[nw * 16 + lm][kb + 1];
            acc = __builtin_amdgcn_wmma_f32_16x16x4_f32(
                false, a, false, b, (short)0, acc, false, false);
            a.x = As1[mw * 16 + lm][kb];
            a.y = As1[mw * 16 + lm][kb + 1];
            b.x = Ws1[nw * 16 + lm][kb];
            b.y = Ws1[nw * 16 + lm][kb + 1];
            acc = __builtin_amdgcn_wmma_f32_16x16x4_f32(
                false, a, false, b, (short)0, acc, false, false);
        }
        __syncthreads();
    }

    const int n  = blockN + nw * 16 + lm;
    const float bv = bias[n];
    const int m0 = blockM + mw * 16 + ((lane >> 4) << 3);
#pragma unroll
    for (int v = 0; v < 8; ++v) {
        float val = acc[v] + bv;
        if (RELU) val = fmaxf(val, 0.0f);
        D[(size_t)(m0 + v) * N + n] = val;
    }
}

// logits = sigmoid(hidden @ wp2^T + bp2) ; one wave32 per link
__global__ __launch_bounds__(256) void link_logit_kernel(const float* __restrict__ hidden,
                                                         const float* __restrict__ wp2,
                                                         const float* __restrict__ bp2,
                                                         float* __restrict__ out) {
    int wid  = blockIdx.x * 8 + (threadIdx.x >> 5);
    int lane = threadIdx.x & 31;
    if (wid >= N_LINKS) return;
    const float* row = hidden + (size_t)wid * HID;
    float s = 0.0f;
#pragma unroll
    for (int k = lane; k < HID; k += 32) s = fmaf(row[k], wp2[k], s);
#pragma unroll
    for (int off = 16; off > 0; off >>= 1) s += __shfl_xor(s, off, 32);
    if (lane == 0) {
        float logit = s + bp2[0];
        out[wid] = 1.0f / (1.0f + expf(-logit));
    }
}

// ---------------------------------------------------------------------------
extern "C" void kernel_launch(void* const* d_in, const int* in_sizes, int n_in,
                              void* d_out, int out_size, void* d_ws, size_t ws_size,
                              hipStream_t stream) {
    (void)in_sizes; (void)n_in; (void)out_size; (void)ws_size;

    const float* x       = (const float*)d_in[0];
    const int*   ei      = (const int*)d_in[1];   // [2, E]
    /* d_in[2] edge_type: result discarded by reference */
    const int*   src_idx = (const int*)d_in[3];
    const int*   dst_idx = (const int*)d_in[4];
    const float* w_emb   = (const float*)d_in[5];
    const float* b_emb   = (const float*)d_in[6];
    /* d_in[7] edge_table, d_in[11,12,16,17] we*,be*: discarded by reference */
    const float* wl0 = (const float*)d_in[8];
    const float* bl0 = (const float*)d_in[9];
    const float* wr0 = (const float*)d_in[10];
    const float* wl1 = (const float*)d_in[13];
    const float* bl1 = (const float*)d_in[14];
    const float* wr1 = (const float*)d_in[15];
    const float* wp1 = (const float*)d_in[18];
    const float* bp1 = (const float*)d_in[19];
    const float* wp2 = (const float*)d_in[20];
    const float* bp2 = (const float*)d_in[21];
    float* out = (float*)d_out;

    // Workspace carve-up (256B aligned)
    char* ws = (char*)d_ws;
    size_t off = 0;
    auto take = [&](size_t bytes) {
        char* p = ws + off;
        off = (off + bytes + 255) & ~(size_t)255;
        return p;
    };
    float* deg = (float*)take((size_t)N_NODES * 4);
    float* B0  = (float*)take((size_t)N_NODES * HID * 4);
    float* B1  = (float*)take((size_t)N_NODES * HID * 4);
    float* B2  = (float*)take((size_t)N_NODES * HID * 4);
    float* comb   = B1;  // 16384*1024 floats (64 MB) aliases B1 (100 MB)
    float* hidden = B2;  // 16384*256  floats aliases B2

    const int T = 256;
    const dim3 gemmGridNodes(N_NODES / 32, HID / 64);   // 3125 x 4
    const dim3 gemmGridLinks(N_LINKS / 32, HID / 64);   //  512 x 4
    const int gElemNodes = (N_NODES * (HID / 4) + T - 1) / T;
    const int gElemEdges = (N_EDGES * (HID / 4) + T - 1) / T;

    // degree
    zero_kernel<<<(N_NODES + T - 1) / T, T, 0, stream>>>(deg, N_NODES);
    degree_kernel<<<(N_EDGES + T - 1) / T, T, 0, stream>>>(ei, deg);

    // h0 = x @ w_emb^T + b_emb
    wmma_gemm_kernel<0><<<gemmGridNodes, T, 0, stream>>>(
        x, w_emb, b_emb, B0, N_NODES, HID, FEAT);

    // ---- SAGE layer 0: h1 = ReLU(mean@wl0^T + bl0 + h0@wr0^T) ----
    zero_kernel<<<(N_NODES * HID + T - 1) / T, T, 0, stream>>>(B1, N_NODES * HID);
    scatter_kernel<<<gElemEdges, T, 0, stream>>>(ei, B0, B1);
    mean_kernel<<<gElemNodes, T, 0, stream>>>(B1, B0, deg);
    wmma_dual_gemm_kernel<1><<<gemmGridNodes, T, 0, stream>>>(
        B1, B0, wl0, wr0, bl0, B2, N_NODES, HID, HID);

    // ---- SAGE layer 1: h2 = ReLU(mean@wl1^T + bl1 + h1@wr1^T) ----
    zero_kernel<<<(N_NODES * HID + T - 1) / T, T, 0, stream>>>(B1, N_NODES * HID);
    scatter_kernel<<<gElemEdges, T, 0, stream>>>(ei, B2, B1);
    mean_kernel<<<gElemNodes, T, 0, stream>>>(B1, B2, deg);
    wmma_dual_gemm_kernel<1><<<gemmGridNodes, T, 0, stream>>>(
        B1, B2, wl1, wr1, bl1, B0, N_NODES, HID, HID);           // h_final in B0

    // ---- link predictor ----
    combined_kernel<<<(N_LINKS * (HID / 4) + T - 1) / T, T, 0, stream>>>(
        B0, src_idx, dst_idx, comb);
    wmma_gemm_kernel<1><<<gemmGridLinks, T, 0, stream>>>(
        comb, wp1, bp1, hidden, N_LINKS, HID, 4 * HID);
    link_logit_kernel<<<N_LINKS / 8, T, 0, stream>>>(hidden, wp2, bp2, out);
}